// NURBSSurfaces_21715354649924
// MI455X (gfx1250) — compile-verified
//
#include <hip/hip_runtime.h>
#include <hip/hip_bf16.h>

// ---------------------------------------------------------------------------
// NURBS surface + normal evaluation for MI455X (gfx1250).
//   - per-(s,f) 12KB control-point tile staged into LDS via the Tensor Data
//     Mover (tensor_load_to_lds, TENSORcnt-fenced)  [CDNA5-specific path]
//   - tile repacked once to float4-padded layout => all 16 stencil taps are
//     16B-aligned ds_load_b128 at immediate offsets from ONE base VGPR
//   - packed-f32 (v_pk_fma_f32) contraction: {value,d/du} accumulated as
//     64-bit register pairs
// ---------------------------------------------------------------------------

#define DEG   3
#define NCP   32
#define NSPAN (NCP - DEG)                 // 29 uniform intervals
#define TILE_FLOATS (NCP * NCP * 3)       // 3072 floats = 12 KB (raw DMA tile)

static constexpr int S_ = 8;
static constexpr int F_ = 4;

typedef unsigned int v4u __attribute__((ext_vector_type(4)));
typedef int          v8i __attribute__((ext_vector_type(8)));
typedef int          v4i __attribute__((ext_vector_type(4)));
typedef float        v2f __attribute__((ext_vector_type(2)));
typedef float        v4f __attribute__((ext_vector_type(4)));

__device__ __forceinline__ float knotv(int idx, float inv) {
  // clamped uniform knot vector: 3 zeros, linspace(0,1,30), 3 ones
  float t = (float)(idx - DEG) * inv;
  return __builtin_fminf(__builtin_fmaxf(t, 0.0f), 1.0f);
}

// NURBS Book A2.3, degree 3, value + first derivative, fully unrolled.
__device__ __forceinline__ void basis_cubic(float u, int span, float inv,
                                            float N[4], float D[4]) {
  const float k0  = knotv(span,     inv);
  const float km1 = knotv(span - 1, inv);
  const float km2 = knotv(span - 2, inv);
  const float k1  = knotv(span + 1, inv);
  const float k2  = knotv(span + 2, inv);
  const float k3  = knotv(span + 3, inv);
  const float l1 = u - k0,  l2 = u - km1, l3 = u - km2;
  const float r1 = k1 - u,  r2 = k2 - u,  r3 = k3 - u;

  // degree 1
  float ndu10 = r1 + l1;
  float t = __builtin_amdgcn_rcpf(ndu10);
  float ndu01 = r1 * t;
  float ndu11 = l1 * t;
  // degree 2
  float ndu20 = r1 + l2;
  t = ndu01 * __builtin_amdgcn_rcpf(ndu20);
  float ndu02 = r1 * t;
  float saved = l2 * t;
  float ndu21 = r2 + l1;
  t = ndu11 * __builtin_amdgcn_rcpf(ndu21);
  float ndu12 = saved + r2 * t;
  float ndu22 = l1 * t;
  // degree 3
  float ndu30 = r1 + l3;
  float rc30 = __builtin_amdgcn_rcpf(ndu30);
  t = ndu02 * rc30;
  N[0] = r1 * t;
  saved = l3 * t;
  float ndu31 = r2 + l2;
  float rc31 = __builtin_amdgcn_rcpf(ndu31);
  t = ndu12 * rc31;
  N[1] = saved + r2 * t;
  saved = l2 * t;
  float ndu32 = r3 + l1;
  float rc32 = __builtin_amdgcn_rcpf(ndu32);
  t = ndu22 * rc32;
  N[2] = saved + r3 * t;
  N[3] = l1 * t;

  // first derivative: N'_r = p*(ndu[r-1][2]/ndu[3][r-1] - ndu[r][2]/ndu[3][r])
  const float q0 = ndu02 * rc30;
  const float q1 = ndu12 * rc31;
  const float q2 = ndu22 * rc32;
  D[0] = -3.0f * q0;
  D[1] =  3.0f * (q0 - q1);
  D[2] =  3.0f * (q1 - q2);
  D[3] =  3.0f * q2;
}

__global__ __launch_bounds__(256)
void nurbs_eval_kernel(const float* __restrict__ ev,
                       const float* __restrict__ cp,
                       float* __restrict__ out_pt,
                       float* __restrict__ out_nrm,
                       int P) {
  __shared__ float lds_raw[TILE_FLOATS];    // 12 KB raw DMA landing zone
  __shared__ v4f   lds_cp4[NCP * NCP];      // 16 KB float4-padded tile

  const int sf = blockIdx.y;                // which (surface, facet)
  const float* cp_tile = cp + (size_t)sf * TILE_FLOATS;

  // ---- TDM DMA: global control-point tile -> LDS (wave 0 issues) ----
  if (threadIdx.x < 32u) {
    // generic LDS pointer: low 32 bits are the LDS byte offset (ISA 10.2)
    const unsigned lds_base = (unsigned)(uintptr_t)(void*)lds_raw;
    const unsigned long long ga = (unsigned long long)(uintptr_t)cp_tile;

    v4u g0;                                  // D# group 0 (128b)
    g0[0] = 1u;                              // count=1 (valid), user mode
    g0[1] = lds_base;                        // lds_addr
    g0[2] = (unsigned)(ga & 0xffffffffu);    // global_addr[31:0]
    g0[3] = (unsigned)((ga >> 32) & 0x01ffffffu) | (2u << 30); // addr[56:32], type=2

    v8i g1;                                  // D# group 1 (256b)
    g1[0] = (2 << 16);                       // data_size=4B, no multicast/pad/iter
    g1[1] = (int)((TILE_FLOATS & 0xffff) << 16);   // tensor_dim0 lo16
    g1[2] = (TILE_FLOATS >> 16) | (1 << 16);       // tensor_dim0 hi16, tensor_dim1=1
    g1[3] = (int)((TILE_FLOATS & 0xffff) << 16);   // tile_dim0 = 3072 (1-D tile)
    g1[4] = 0;                               // tile_dim1=0, tile_dim2=0
    g1[5] = TILE_FLOATS;                     // tensor_dim0_stride lo32
    g1[6] = 0;
    g1[7] = 0;

    v4i g2 = {0, 0, 0, 0};                   // 1-D tensor: group 2 unused
    v4i g3 = {0, 0, 0, 0};                   // group 3 unused
    v8i g4 = {0, 0, 0, 0, 0, 0, 0, 0};       // 6-arg toolchain variant: extra group

    __builtin_amdgcn_tensor_load_to_lds(g0, g1, g2, g3, g4, 0);
    __builtin_amdgcn_s_wait_tensorcnt(0);    // TENSORcnt == 0 -> tile resident
  }
  __syncthreads();                           // raw tile visible to all 8 waves

  // ---- one-time repack: AoS float[3] -> float4 (16B-aligned taps) ----
  for (int e = (int)threadIdx.x; e < NCP * NCP; e += (int)blockDim.x) {
    v4f t;
    t.x = lds_raw[3 * e + 0];
    t.y = lds_raw[3 * e + 1];
    t.z = lds_raw[3 * e + 2];
    t.w = 0.0f;
    lds_cp4[e] = t;
  }
  __syncthreads();

  const float inv = 1.0f / (float)NSPAN;
  const size_t base   = (size_t)sf * (size_t)P;
  const int    stride = gridDim.x * blockDim.x;

  for (int p = blockIdx.x * blockDim.x + (int)threadIdx.x; p < P; p += stride) {
    const float2 uv = ((const float2*)ev)[base + (size_t)p];
    const float u = uv.x, v = uv.y;

    const int span_u = (int)(u * (float)NSPAN) + DEG;   // u,v in [0,1)
    const int span_v = (int)(v * (float)NSPAN) + DEG;

    float Nu[4], Du[4], Nv[4], Dv[4];
    basis_cubic(u, span_u, inv, Nu, Du);
    basis_cubic(v, span_v, inv, Nv, Dv);

    // single base index; all 16 taps at immediate offsets (i*32+j)*16 bytes
    const int base_e = (span_u - DEG) * NCP + (span_v - DEG);
    const v4f* __restrict__ tap = &lds_cp4[base_e];

    // stage 1 (contract over i): pairs {T0,T1} = {sum Nu*cp, sum Du*cp}
    v2f Tx[4] = {};  // per j: {T0x, T1x}
    v2f Ty[4] = {};
    v2f Tz[4] = {};
#pragma unroll
    for (int i = 0; i < 4; ++i) {
      const v2f w = {Nu[i], Du[i]};           // packed weight pair
#pragma unroll
      for (int j = 0; j < 4; ++j) {
        const v4f c = tap[i * NCP + j];       // ds_load_b128, immediate offset
        Tx[j] = __builtin_elementwise_fma(w, c.xx, Tx[j]);  // v_pk_fma_f32
        Ty[j] = __builtin_elementwise_fma(w, c.yy, Ty[j]);
        Tz[j] = __builtin_elementwise_fma(w, c.zz, Tz[j]);
      }
    }

    // stage 2 (contract over j): {pt,du} packed; dv scalar off the lo halves
    v2f PDx = {0.f, 0.f}, PDy = {0.f, 0.f}, PDz = {0.f, 0.f};
    float dvx = 0.f, dvy = 0.f, dvz = 0.f;
#pragma unroll
    for (int j = 0; j < 4; ++j) {
      const v2f nv = {Nv[j], Nv[j]};
      PDx = __builtin_elementwise_fma(nv, Tx[j], PDx);
      PDy = __builtin_elementwise_fma(nv, Ty[j], PDy);
      PDz = __builtin_elementwise_fma(nv, Tz[j], PDz);
      dvx = fmaf(Dv[j], Tx[j].x, dvx);
      dvy = fmaf(Dv[j], Ty[j].x, dvy);
      dvz = fmaf(Dv[j], Tz[j].x, dvz);
    }

    const float dux = PDx.y, duy = PDy.y, duz = PDz.y;   // hi halves: d/du
    const float nx = duy * dvz - duz * dvy;
    const float ny = duz * dvx - dux * dvz;
    const float nz = dux * dvy - duy * dvx;
    const float len  = sqrtf(fmaf(nx, nx, fmaf(ny, ny, nz * nz)));
    const float rinv = __builtin_amdgcn_rcpf(len + 1e-12f);

    const size_t o = (base + (size_t)p) * 3;
    out_pt[o + 0] = PDx.x;                                // lo halves: point
    out_pt[o + 1] = PDy.x;
    out_pt[o + 2] = PDz.x;
    out_nrm[o + 0] = nx * rinv;
    out_nrm[o + 1] = ny * rinv;
    out_nrm[o + 2] = nz * rinv;
  }
}

extern "C" void kernel_launch(void* const* d_in, const int* in_sizes, int n_in,
                              void* d_out, int out_size, void* d_ws, size_t ws_size,
                              hipStream_t stream) {
  const float* ev = (const float*)d_in[0];   // [S,F,P,2] f32
  const float* cp = (const float*)d_in[1];   // [S,F,32,32,3] f32

  const int SF = S_ * F_;
  const int P  = in_sizes[0] / (SF * 2);     // 200000

  float* out_pt  = (float*)d_out;                          // [S,F,P,3]
  float* out_nrm = out_pt + (size_t)SF * (size_t)P * 3;    // [S,F,P,3]

  dim3 grid(40, SF);   // 1280 blocks of 8 wave32s; DMA amortized over ~5k pts
  dim3 block(256);
  nurbs_eval_kernel<<<grid, block, 0, stream>>>(ev, cp, out_pt, out_nrm, P);
}